// GivensRotationPerHead_2860448219294
// MI455X (gfx1250) — compile-verified
//
#include <hip/hip_runtime.h>
#include <hip/hip_bf16.h>

#define DIM 128
#define NTHETA 8128        // C(128,2)
#define NCHUNK 4
#define SLABW 32           // columns per combine block

typedef __attribute__((ext_vector_type(2))) float v2f;
typedef __attribute__((ext_vector_type(8))) float v8f;

// -------------------------------------------------------------------------
// Kernel A: apply one chunk of Givens rotations (i in [i0,i1)) to identity.
// Grid: (K, 64 heads, 2 column-halves). Block: 64 threads, each owns 1 column.
// State: LDS 128 rows x 64 cols (32KB). Pivot row element qi lives in a VGPR
// for the whole i-run, so the serial dependency chain is VALU-only; the qj
// LDS load of each iteration is independent and pipelines.
// -------------------------------------------------------------------------
__global__ __launch_bounds__(64) void givens_chunk(
    const float* __restrict__ thetas, float* __restrict__ dst, int K)
{
    extern __shared__ float smem[];
    float* S   = smem;               // [128][64], column-major ownership
    float* csb = smem + DIM * 64;    // [2*128] interleaved (c,s) pairs

    const int t     = threadIdx.x;           // 0..63
    const int chunk = blockIdx.x;
    const int head  = blockIdx.y;
    const int col   = blockIdx.z * 64 + t;   // global column 0..127

    int i0, i1;
    if (K == NCHUNK) {
        const int b[NCHUNK + 1] = {0, 17, 37, 64, 128};  // balanced rotation counts
        i0 = b[chunk]; i1 = b[chunk + 1];
    } else { i0 = 0; i1 = DIM; }

    const float* th  = thetas + (size_t)head * NTHETA;
    float*       out = dst + ((size_t)head * K + chunk) * (DIM * DIM);

    // Identity init: each thread touches only its own column -> no sync needed.
    #pragma unroll 4
    for (int r = 0; r < DIM; ++r) S[r * 64 + t] = (r == col) ? 1.0f : 0.0f;

    for (int i = i0; i < i1; ++i) {
        const int len  = DIM - 1 - i;             // rotations (i, i+1..127)
        const int base = i * (2 * DIM - 1 - i) / 2;  // triu flat index of (i, i+1)

        // Cooperative cos/sin for this run into LDS broadcast buffer.
        for (int u = t; u < len; u += 64) {
            float sv, cv;
            __sincosf(th[base + u], &sv, &cv);
            csb[2 * u]     = cv;
            csb[2 * u + 1] = sv;
        }
        __syncthreads();

        float qi = S[i * 64 + t];                 // pivot element in a VGPR
        for (int jj = 0; jj < len; ++jj) {
            const float c  = csb[2 * jj];
            const float s  = csb[2 * jj + 1];
            const int   j  = i + 1 + jj;
            const float qj = S[j * 64 + t];
            const float nqj = fmaf(c, qj, -s * qi);   // uses old qi
            qi = fmaf(c, qi, s * qj);
            S[j * 64 + t] = nqj;
        }
        S[i * 64 + t] = qi;
        __syncthreads();   // protect csb before next run overwrites it
    }

    // Write chunk matrix (row-major 128x128); lanes -> consecutive addresses.
    #pragma unroll 4
    for (int r = 0; r < DIM; ++r) out[r * DIM + col] = S[r * 64 + t];
}

// -------------------------------------------------------------------------
// Kernel B: per head, Q = C3*C2*C1*C0 via chained WMMA f32 16x16x4 GEMMs.
// Grid: (64 heads, 4 column-slabs of 32). Block: 256 threads (8 waves).
// acc/nxt: double-buffered 128x32 slabs in LDS (32KB). A-fragments stream
// from global (hot in L2: 16MB of chunk data vs 192MB L2).
//
// ISA 7.12.2 operand packing (wave32):
//   A 16x4 : lane = m + 16*(k>>1), vgpr = k&1
//   B 4x16 : lane = n + 16*(k>>1), vgpr = k&1
//   D 16x16: vgpr r -> row r (lanes 0-15) / row r+8 (lanes 16-31), col = lane%16
// -------------------------------------------------------------------------
__global__ __launch_bounds__(256) void combine_gemm(
    const float* __restrict__ chunks, float* __restrict__ outp)
{
    extern __shared__ float smem[];
    float* acc = smem;                 // [128][32]
    float* nxt = smem + DIM * SLABW;   // [128][32]

    const int t    = threadIdx.x;
    const int head = blockIdx.x;
    const int col0 = blockIdx.y * SLABW;
    const float* src = chunks + (size_t)head * NCHUNK * DIM * DIM;

    // acc <- C0 slab (float4, coalesced)
    for (int i = t; i < DIM * SLABW / 4; i += 256) {
        const int row = i >> 3;            // 8 float4 per 32-col row
        const int c4  = (i & 7) << 2;
        *(float4*)&acc[row * SLABW + c4] =
            *(const float4*)&src[row * DIM + col0 + c4];
    }
    __syncthreads();

    const int wave = t >> 5, lane = t & 31;
    const int half = lane >> 4, l16 = lane & 15;
    const int m = wave;                    // tile row 0..7

    for (int ci = 1; ci < NCHUNK; ++ci) {
        const float* Cm = src + (size_t)ci * DIM * DIM;   // A matrix (128x128)
        #pragma unroll
        for (int n = 0; n < SLABW / 16; ++n) {            // tile cols
            v8f d = {0.f, 0.f, 0.f, 0.f, 0.f, 0.f, 0.f, 0.f};
            for (int k0 = 0; k0 < DIM; k0 += 4) {
                const int kb = k0 + half * 2;
                const float* ap = Cm + (m * 16 + l16) * DIM + kb;
                v2f a; a.x = ap[0]; a.y = ap[1];                       // A[m][k], A[m][k+1]
                v2f b; b.x = acc[kb * SLABW + n * 16 + l16];           // B[k][n]
                       b.y = acc[(kb + 1) * SLABW + n * 16 + l16];     // B[k+1][n]
                d = __builtin_amdgcn_wmma_f32_16x16x4_f32(
                        /*neg_a=*/false, a, /*neg_b=*/false, b,
                        /*c_mod=*/(short)0, d, /*reuse_a=*/false, /*reuse_b=*/false);
            }
            #pragma unroll
            for (int r = 0; r < 8; ++r)
                nxt[(m * 16 + r + half * 8) * SLABW + n * 16 + l16] = d[r];
        }
        __syncthreads();                   // all reads of acc done, nxt complete
        float* tp = acc; acc = nxt; nxt = tp;
    }

    // Final slab -> d_out
    float* outHead = outp + (size_t)head * DIM * DIM;
    for (int i = t; i < DIM * SLABW / 4; i += 256) {
        const int row = i >> 3;
        const int c4  = (i & 7) << 2;
        *(float4*)&outHead[row * DIM + col0 + c4] =
            *(const float4*)&acc[row * SLABW + c4];
    }
}

// -------------------------------------------------------------------------
extern "C" void kernel_launch(void* const* d_in, const int* in_sizes, int n_in,
                              void* d_out, int out_size, void* d_ws, size_t ws_size,
                              hipStream_t stream) {
    const float* thetas = (const float*)d_in[0];   // (64, 8128) f32
    // d_in[1]/d_in[2] (rot_i/rot_j) are the canonical triu(128,1) indices; the
    // scan order is reproduced analytically, so they are not needed on device.
    float* out = (float*)d_out;                    // (64, 128, 128) f32

    const size_t need = (size_t)64 * NCHUNK * DIM * DIM * sizeof(float); // 16 MB
    const size_t smemA = (DIM * 64 + 2 * DIM) * sizeof(float);           // 33 KB
    const size_t smemB = (size_t)2 * DIM * SLABW * sizeof(float);        // 32 KB

    if (ws_size >= need) {
        // Phase 1: 4 independent rotation chunks per head (4x shorter chain).
        dim3 gA(NCHUNK, 64, 2);
        givens_chunk<<<gA, 64, smemA, stream>>>(thetas, (float*)d_ws, NCHUNK);
        // Phase 2: Q = C3*C2*C1*C0 with WMMA f32 16x16x4.
        dim3 gB(64, DIM / SLABW);
        combine_gemm<<<gB, 256, smemB, stream>>>((const float*)d_ws, out);
    } else {
        // Fallback: single full-range chunk straight into d_out.
        dim3 gA(1, 64, 2);
        givens_chunk<<<gA, 64, smemA, stream>>>(thetas, out, 1);
    }
}